// CausalNet_69870527971800
// MI455X (gfx1250) — compile-verified
//
#include <hip/hip_runtime.h>
#include <cstdint>

// Problem constants (fixed by the reference setup).
#define T_LEN 256
#define NB    16           // number of options b
#define NEGV  (-1000000000.0f)

typedef unsigned int u32x4 __attribute__((ext_vector_type(4)));
typedef int          i32x4 __attribute__((ext_vector_type(4)));
typedef int          i32x8 __attribute__((ext_vector_type(8)));

// ---------------------------------------------------------------------------
// Helpers
// ---------------------------------------------------------------------------
__device__ __forceinline__ float lse2(float x, float y) {
  float m = fmaxf(x, y);
  return m + __logf(__expf(x - m) + __expf(y - m));
}

// Block-wide max over 256 threads (8 wave32). 'red' is 8 floats of LDS.
__device__ __forceinline__ float block_max(float v, volatile float* red, int tid) {
#pragma unroll
  for (int m = 16; m >= 1; m >>= 1) v = fmaxf(v, __shfl_xor(v, m, 32));
  if ((tid & 31) == 0) red[tid >> 5] = v;
  __syncthreads();
  if (tid == 0) {
    float r = red[0];
#pragma unroll
    for (int i = 1; i < 8; ++i) r = fmaxf(r, red[i]);
    red[0] = r;
  }
  __syncthreads();
  float r = red[0];
  __syncthreads();
  return r;
}

__device__ __forceinline__ float block_sum(float v, volatile float* red, int tid) {
#pragma unroll
  for (int m = 16; m >= 1; m >>= 1) v += __shfl_xor(v, m, 32);
  if ((tid & 31) == 0) red[tid >> 5] = v;
  __syncthreads();
  if (tid == 0) {
    float r = 0.0f;
#pragma unroll
    for (int i = 0; i < 8; ++i) r += red[i];
    red[0] = r;
  }
  __syncthreads();
  float r = red[0];
  __syncthreads();
  return r;
}

// ---------------------------------------------------------------------------
// TDM: DMA a 2-D tile  pen[s = 0..rows-1][b = 0..15]  (rows x 64B, row stride
// (T+1)*16 floats) from global memory into LDS.  Issued by one wave; tracked
// with TENSORcnt.  Descriptor layout per CDNA5 ISA §8 (D# groups 0/1).
// ---------------------------------------------------------------------------
__device__ __forceinline__ void tdm_load_pen_tile(const float* tile_base, int rows,
                                                  unsigned lds_byte_off) {
  unsigned long long ga = (unsigned long long)(uintptr_t)tile_base;
  u32x4 g0 = {
      1u,                                                // [1:0] count=1, user D#
      lds_byte_off,                                      // [63:32] lds_addr (bytes)
      (unsigned)(ga & 0xffffffffu),                      // [95:64] global_addr lo
      ((unsigned)((ga >> 32) & 0x01ffffffu)) | (2u << 30) // [120:96] addr hi, [127:126] type=2
  };
  i32x8 g1 = {
      (int)(2u << 16),                          // data_size = 2 (4 bytes)
      (int)(16u << 16),                         // tensor_dim0 = 16 (bits 63:48)
      (int)(((unsigned)T_LEN & 0xffffu) << 16), // tensor_dim1 = 256 (bits 95:80)
      (int)(16u << 16),                         // tile_dim0 = 16 (bits 127:112)
      (int)(rows & 0xffff),                     // tile_dim1 = rows (bits 143:128)
      (int)((T_LEN + 1) * NB),                  // tensor_dim0_stride = 4112 elems
      0, 0                                      // dim1 stride unused (2-D tile)
  };
  i32x4 z4 = {0, 0, 0, 0};
#if __clang_major__ >= 23
  i32x8 z8 = {0, 0, 0, 0, 0, 0, 0, 0};
  __builtin_amdgcn_tensor_load_to_lds(g0, g1, z4, z4, z8, 0);
#else
  __builtin_amdgcn_tensor_load_to_lds(g0, g1, z4, z4, 0);
#endif
}

// ---------------------------------------------------------------------------
// Kernel 1: backward scan.  One wave per batch; lane -> option b (lanes 16-31
// mirror 0-15 so the width-16 shuffle reductions stay convergent).
// Stores bw[B][t][b] = BW[t][b][1] for t=0..T-1 and bw[B][T][b] = 0.
// ---------------------------------------------------------------------------
__global__ __launch_bounds__(32) void bwd_kernel(
    const float* __restrict__ alp,    // (B,T,16)
    const float* __restrict__ slp,    // (B,T+1,16,2)
    const float* __restrict__ stlp,   // (B,T+1,16)
    const float* __restrict__ masks,  // (B,T)
    float* __restrict__ bw)           // (B,T+1,16)
{
  const int Bi   = blockIdx.x;
  const int lane = threadIdx.x;
  const int b    = lane & 15;

  const float* alpB   = alp   + (size_t)Bi * T_LEN * NB;
  const float* slpB   = slp   + (size_t)Bi * (T_LEN + 1) * NB * 2;
  const float* stlpB  = stlp  + (size_t)Bi * (T_LEN + 1) * NB;
  const float* masksB = masks + (size_t)Bi * T_LEN;
  float*       bwB    = bw    + (size_t)Bi * (T_LEN + 1) * NB;

  float f0 = NEGV;  // f_next[b][0]
  float f1 = 0.0f;  // f_next[b][1]
  if (lane < NB) bwB[T_LEN * NB + b] = 0.0f;  // BW_full[T] slot (fT[...,1] = 0)

  for (int t = T_LEN - 1; t >= 0; --t) {
    float a   = alpB[t * NB + b];
    float st0 = slpB[((t + 1) * NB + b) * 2 + 1];  // reversed e=0 -> orig 1
    float st1 = slpB[((t + 1) * NB + b) * 2 + 0];  // reversed e=1 -> orig 0
    float stt = stlpB[t * NB + b] - 0.5f;          // ABSTRACT_PEN
    float m   = masksB[t];

    float f0c = a + lse2(st0 + f0, st1 + f1);
    float x   = stt + f0c;                          // f1c = LSE_b(x)
    float mx  = x;
#pragma unroll
    for (int d = 8; d >= 1; d >>= 1) mx = fmaxf(mx, __shfl_xor(mx, d, 16));
    float e = __expf(x - mx);
#pragma unroll
    for (int d = 8; d >= 1; d >>= 1) e += __shfl_xor(e, d, 16);
    float f1c = mx + __logf(e);

    bool act = (m > 0.0f);
    f0 = act ? f0c : NEGV;
    f1 = act ? f1c : 0.0f;
    if (lane < NB) bwB[t * NB + b] = f1;
  }
}

// ---------------------------------------------------------------------------
// Kernel 2: two-pass forward scan per batch (256 threads = 8 waves).
// State g[s][b] lives in LDS; thread (sg = tid>>4, b = tid&15) owns slots
// s in {sg, sg+16, ...}.  Pass A: per-step LSE -> S_t history and logZ = S_L.
// Pass B: replay reduction-free; TDM streams pen tiles into a double buffer;
// accumulate cc = sum exp(f1 + BW - logZ) * pen in registers.
// ---------------------------------------------------------------------------
__global__ __launch_bounds__(256) void fwd_kernel(
    const float* __restrict__ alp,      // (B,T,16)
    const float* __restrict__ slp,      // (B,T+1,16,2)
    const float* __restrict__ stlp,     // (B,T+1,16)
    const float* __restrict__ pen,      // (B,T+1,T+1,16)
    const int*   __restrict__ lengths,  // (B,)
    const float* __restrict__ bw,       // (B,T+1,16)
    float* __restrict__ logp_out,       // (B,)
    float* __restrict__ cc_out)         // (B,)
{
  __shared__ float g[T_LEN * NB];           // 16 KB
  __shared__ float Sh[T_LEN + 1];           // S_t history
  __shared__ float addv[NB], s1v[NB], spa[NB], wv[NB], ls0[NB];
  __shared__ float red[8];
  __shared__ float penbuf[2][T_LEN * NB];   // 2 x 16 KB double buffer

  const int tid = threadIdx.x;
  const int Bi  = blockIdx.x;
  const int b   = tid & 15;
  const int sg  = tid >> 4;
  const int L   = lengths[Bi];

  const float* alpB  = alp  + (size_t)Bi * T_LEN * NB;
  const float* slpB  = slp  + (size_t)Bi * (T_LEN + 1) * NB * 2;
  const float* stlpB = stlp + (size_t)Bi * (T_LEN + 1) * NB;
  const float* bwB   = bw   + (size_t)Bi * (T_LEN + 1) * NB;
  const float* penB  = pen  + (size_t)Bi * (T_LEN + 1) * (T_LEN + 1) * NB;

  if (tid == 0) Sh[0] = 0.0f;     // LSE of f0[...,1] (single 0 entry)
  if (tid < NB) ls0[tid] = 0.0f;  // stop0 of "step 0" (unused at t=1)
  __syncthreads();

  // ---------------- Pass A: S_t history + logZ ----------------
  for (int t = 1; t <= L; ++t) {
    if (tid < NB) {
      float a   = alpB[(t - 1) * NB + tid];
      float st0 = slpB[(t * NB + tid) * 2 + 1];
      float st1 = slpB[(t * NB + tid) * 2 + 0];
      float stt = stlpB[(t - 1) * NB + tid] - 0.5f;
      addv[tid] = ls0[tid] + a;   // stop0_{t-1} + a_t (continuation add)
      spa[tid]  = stt + a;        // start_t + a_t    (new-segment add)
      s1v[tid]  = st1;
      ls0[tid]  = st0;
    }
    __syncthreads();

    float av = addv[b], s1 = s1v[b], seg = Sh[t - 1];
    float vmax = -INFINITY;
    float vals[16];
#pragma unroll
    for (int k = 0; k < 16; ++k) {
      int s = (k << 4) + sg;
      float f1 = -INFINITY;
      if (s < t) {
        float gv = (s == t - 1) ? (seg + spa[b]) : (g[s * NB + b] + av);
        g[s * NB + b] = gv;
        f1 = gv + s1;
      }
      vals[k] = f1;
      vmax = fmaxf(vmax, f1);
    }
    float M = block_max(vmax, red, tid);
    float lsum = 0.0f;
#pragma unroll
    for (int k = 0; k < 16; ++k) lsum += __expf(vals[k] - M);
    float tot = block_sum(lsum, red, tid);
    if (tid == 0) Sh[t] = M + __logf(tot);
    // next iteration's constants-barrier publishes Sh[t]
  }
  __syncthreads();
  const float logZ = Sh[L];       // total_logp for this batch
  if (tid == 0) logp_out[Bi] = logZ;
  if (tid < NB) ls0[tid] = 0.0f;
  __syncthreads();

  // ---------------- Pass B: replay + cc accumulation ----------------
  float acc = 0.0f;
  if (tid == 0) {
    // Prologue: DMA tile for t=1 (rows s<1) into buffer 1.
    tdm_load_pen_tile(penB + (size_t)1 * NB, 1,
                      (unsigned)(uintptr_t)&penbuf[1][0]);
  }
  for (int t = 1; t <= L; ++t) {
    if (tid == 0 && t < L) {
      // Prefetch next tile (rows s<t+1) into the other buffer.
      tdm_load_pen_tile(penB + (size_t)(t + 1) * NB, t + 1,
                        (unsigned)(uintptr_t)&penbuf[(t + 1) & 1][0]);
    }
    if (tid < NB) {
      float a   = alpB[(t - 1) * NB + tid];
      float st0 = slpB[(t * NB + tid) * 2 + 1];
      float st1 = slpB[(t * NB + tid) * 2 + 0];
      float stt = stlpB[(t - 1) * NB + tid] - 0.5f;
      addv[tid] = ls0[tid] + a;
      spa[tid]  = stt + a;
      wv[tid]   = st1 + bwB[t * NB + tid] - logZ;  // stop1 + BW_full[t] - logZ
      ls0[tid]  = st0;
    }
    if (tid == 0) {
      if (t < L) __builtin_amdgcn_s_wait_tensorcnt(1);  // tile t landed
      else       __builtin_amdgcn_s_wait_tensorcnt(0);
    }
    __syncthreads();  // publish constants + TDM-written LDS tile

    float av = addv[b], w = wv[b], seg = Sh[t - 1];
    const float* pb = &penbuf[t & 1][0];
#pragma unroll
    for (int k = 0; k < 16; ++k) {
      int s = (k << 4) + sg;
      if (s < t) {
        float gv = (s == t - 1) ? (seg + spa[b]) : (g[s * NB + b] + av);
        g[s * NB + b] = gv;
        acc += __expf(gv + w) * pb[s * NB + b];   // posterior * causal_pen
      }
    }
    __syncthreads();  // all readers done before buffer parity is reused
  }
  float cc = block_sum(acc, red, tid);
  if (tid == 0) cc_out[Bi] = cc;
}

// ---------------------------------------------------------------------------
// Kernel 3: deterministic final reduction over batches.
// ---------------------------------------------------------------------------
__global__ void finalize_kernel(const float* __restrict__ logp,
                                const float* __restrict__ cc,
                                float* __restrict__ out, int B) {
  if (threadIdx.x == 0 && blockIdx.x == 0) {
    float s0 = 0.0f, s1 = 0.0f;
    for (int i = 0; i < B; ++i) { s0 += logp[i]; s1 += cc[i]; }
    out[0] = s0;  // sum(total_logp)
    out[1] = s1;  // total_cc
  }
}

// ---------------------------------------------------------------------------
// Host entry
// ---------------------------------------------------------------------------
extern "C" void kernel_launch(void* const* d_in, const int* in_sizes, int n_in,
                              void* d_out, int out_size, void* d_ws, size_t ws_size,
                              hipStream_t stream) {
  (void)n_in; (void)out_size; (void)ws_size;
  const float* alp   = (const float*)d_in[0];  // action_logps (B,T,16)
  const float* slp   = (const float*)d_in[1];  // stop_logps   (B,T+1,16,2)
  const float* stlp  = (const float*)d_in[2];  // start_logps  (B,T+1,16)
  const float* pen   = (const float*)d_in[3];  // causal_pens  (B,T+1,T+1,16)
  const float* masks = (const float*)d_in[4];  // masks        (B,T)
  const int*   lens  = (const int*)d_in[5];    // lengths      (B,)
  const int B = in_sizes[5];

  float* ws   = (float*)d_ws;
  float* bw   = ws;                                       // B*(T+1)*16
  float* logp = bw + (size_t)B * (T_LEN + 1) * NB;        // B
  float* cc   = logp + B;                                 // B

  bwd_kernel<<<B, 32, 0, stream>>>(alp, slp, stlp, masks, bw);
  fwd_kernel<<<B, 256, 0, stream>>>(alp, slp, stlp, pen, lens, bw, logp, cc);
  finalize_kernel<<<1, 32, 0, stream>>>(logp, cc, (float*)d_out, B);
}